// RayClsSample_8727373546151
// MI455X (gfx1250) — compile-verified
//
#include <hip/hip_runtime.h>
#include <stdint.h>

// Problem shape (from reference): B=16, N=NP=4096, F=64, C=68, n_total=8192, m=N.
// Sampler: one workgroup per batch, 1024 threads (32 wave32s), 8 points/thread.

#define BLK   1024
#define NTOT  8192
#define PPT   (NTOT / BLK)   // points per thread = 8
#define NWAVE (BLK / 32)     // 32 wave32s

// ---- CDNA5 async global->LDS staging (per-lane, ASYNCcnt-tracked) ----------
__device__ __forceinline__ void async_copy_b128(const float* gsrc, float* lds_dst) {
    unsigned lds_addr = (unsigned)(uintptr_t)lds_dst;        // addrspace(3) byte offset
    unsigned long long gaddr = (unsigned long long)(uintptr_t)gsrc;
    asm volatile("global_load_async_to_lds_b128 %0, %1, off"
                 :: "v"(lds_addr), "v"(gaddr) : "memory");
}
__device__ __forceinline__ void wait_async0() {
    asm volatile("s_wait_asynccnt 0" ::: "memory");
}

__device__ __forceinline__ unsigned long long wave_min_u64(unsigned long long v) {
    #pragma unroll
    for (int off = 16; off > 0; off >>= 1) {
        const unsigned long long o = __shfl_xor(v, off, 32);
        v = (o < v) ? o : v;
    }
    return v;
}

// ---- Kernel 1: sequential minimum-density sampling (one block per batch) ---
__global__ __launch_bounds__(BLK)
void mds_sample_kernel(const float* __restrict__ partial,   // [B,3,NP]
                       const float* __restrict__ outs,      // [B,3,N]
                       const float* __restrict__ sigma,     // [B]
                       int N, int NP, int m,
                       int* __restrict__ idx_out)           // [B,m]
{
    __shared__ float sx[NTOT], sy[NTOT], sz[NTOT];
    __shared__ unsigned long long red[2][NWAVE];  // double-buffered by step parity

    const int b    = blockIdx.x;
    const int t    = threadIdx.x;
    const int lane = t & 31;
    const int wid  = t >> 5;

    // ---- Stage xyz into LDS: row c of `outs` -> [0,N), row c of `partial` -> [N,N+NP)
    float* dst[3] = { sx, sy, sz };
    for (int c = 0; c < 3; ++c) {
        const float* srcN = outs    + ((size_t)b * 3 + c) * (size_t)N;
        for (int i = t; i < N / 4; i += BLK)
            async_copy_b128(srcN + 4 * i, dst[c] + 4 * i);
        const float* srcP = partial + ((size_t)b * 3 + c) * (size_t)NP;
        for (int i = t; i < NP / 4; i += BLK)
            async_copy_b128(srcP + 4 * i, dst[c] + N + 4 * i);
    }
    wait_async0();
    __syncthreads();

    // ---- Thread-private state: coords + density in registers (no LDS in hot loop)
    float mx[PPT], my[PPT], mz[PPT], dens[PPT];
    #pragma unroll
    for (int k = 0; k < PPT; ++k) {
        const int i = t + k * BLK;
        mx[k] = sx[i]; my[k] = sy[i]; mz[k] = sz[i];
        dens[k] = 0.0f;
    }

    const float s   = sigma[b];
    const float inv = 1.0f / (2.0f * s * s);

    int last = 0;
    for (int step = 0; step < m; ++step) {
        if (t == 0) idx_out[(size_t)b * m + step] = last;
        if (step + 1 == m) break;   // final argmin is discarded by the scan

        const float px = sx[last], py = sy[last], pz = sz[last];

        // Density update + local argmin with packed (value,index) key.
        // Densities are >= 0 => float bits are order-monotonic as unsigned,
        // and min over (bits<<32)|idx == first-index argmin (matches jnp.argmin).
        unsigned long long best = ~0ull;
        #pragma unroll
        for (int k = 0; k < PPT; ++k) {
            const float dx = mx[k] - px, dy = my[k] - py, dz = mz[k] - pz;
            const float d2 = dx * dx + dy * dy + dz * dz;
            dens[k] += __expf(-d2 * inv);
            const unsigned long long key =
                ((unsigned long long)__float_as_uint(dens[k]) << 32)
                | (unsigned)(t + k * BLK);
            best = (key < best) ? key : best;
        }

        // Stage 1: wave32 shuffle reduction; leaders publish per-wave winner.
        best = wave_min_u64(best);
        const int buf = step & 1;
        if (lane == 0) red[buf][wid] = best;
        __syncthreads();

        // Stage 2: EVERY wave redundantly reduces the 32 winners -> all threads
        // hold the block argmin with no second barrier / broadcast round-trip.
        unsigned long long v = red[buf][lane];
        v = wave_min_u64(v);
        last = (int)(unsigned)(v & 0xffffffffu);
    }
}

// ---- Kernel 2: channel gather out[b,c,j] = xx[b,c,idx[b,j]] ----------------
__global__ void gather_kernel(const float* __restrict__ partial,  // [B,3,NP]
                              const float* __restrict__ outs,     // [B,3,N]
                              const float* __restrict__ feat,     // [B,F,N]
                              const int*   __restrict__ idx,      // [B,m]
                              float* __restrict__ out,            // [B,C,m]
                              int B_, int N, int NP, int F_, int m)
{
    const int C = 3 + F_ + 1;
    const long long total = (long long)B_ * C * m;
    const long long gid = (long long)blockIdx.x * blockDim.x + threadIdx.x;
    if (gid >= total) return;

    const int j = (int)(gid % m);
    const int c = (int)((gid / m) % C);
    const int b = (int)(gid / ((long long)m * C));
    const int col = idx[(size_t)b * m + j];

    float v;
    if (c < 3) {
        v = (col < N) ? outs[((size_t)b * 3 + c) * N + col]
                      : partial[((size_t)b * 3 + c) * NP + (col - N)];
    } else if (c < 3 + F_) {
        v = (col < N) ? feat[((size_t)b * F_ + (c - 3)) * N + col] : 0.0f;
    } else {
        v = (col < N) ? 1.0f : 0.0f;
    }
    out[gid] = v;
}

extern "C" void kernel_launch(void* const* d_in, const int* in_sizes, int n_in,
                              void* d_out, int out_size, void* d_ws, size_t ws_size,
                              hipStream_t stream) {
    const float* partial = (const float*)d_in[0];   // [B,3,NP]
    const float* outs    = (const float*)d_in[1];   // [B,3,N]
    const float* feat    = (const float*)d_in[2];   // [B,F,N]
    const float* sigma   = (const float*)d_in[3];   // [B]

    const int B_  = in_sizes[3];
    const int NP_ = in_sizes[0] / (3 * B_);
    const int N_  = in_sizes[1] / (3 * B_);
    const int F_  = in_sizes[2] / (B_ * N_);
    const int m   = N_;

    int* idx = (int*)d_ws;  // B*m int32

    mds_sample_kernel<<<B_, BLK, 0, stream>>>(partial, outs, sigma, N_, NP_, m, idx);

    const long long total = (long long)B_ * (3 + F_ + 1) * m;
    const int threads = 256;
    const int blocks = (int)((total + threads - 1) / threads);
    gather_kernel<<<blocks, threads, 0, stream>>>(partial, outs, feat, idx,
                                                  (float*)d_out, B_, N_, NP_, F_, m);
}